// GaussianState_24318104830346
// MI455X (gfx1250) — compile-verified
//
#include <hip/hip_runtime.h>
#include <math.h>

typedef __attribute__((ext_vector_type(2))) float v2f;
typedef __attribute__((ext_vector_type(8))) float v8f;

#if defined(__has_builtin)
#if __has_builtin(__builtin_amdgcn_tensor_load_to_lds)
#define HAVE_TDM 1
#endif
#endif

namespace gs {
constexpr int n = 64;     // spin sites
constexpr int N = 128;    // Majorana modes
constexpr int B = 128;    // batch

// workspace layout (float offsets)
constexpr size_t OFF_H    = 0;                                   // 2 * 128*128
constexpr size_t OFF_P2   = OFF_H    + (size_t)2*N*N;
constexpr size_t OFF_P3   = OFF_P2   + (size_t)2*N*N;
constexpr size_t OFF_P5   = OFF_P3   + (size_t)2*N*N;
constexpr size_t OFF_G    = OFF_P5   + (size_t)2*N*N;
constexpr size_t OFF_TR   = OFF_G    + (size_t)2*N*N;            // tr2,tr4,tr6 per g
constexpr size_t OFF_LT   = OFF_TR   + 8;                        // lt per g
constexpr size_t OFF_A    = OFF_LT   + 8;                        // 2 * [re 4096 | im 4096]
constexpr size_t OFF_AINV = OFF_A    + (size_t)2*2*n*n;
constexpr size_t OFF_PFA  = OFF_AINV + (size_t)2*2*n*n;          // 2 * {sre,sim,log,pad}
constexpr size_t OFF_V    = OFF_PFA  + 8;                        // B * [re 8192 | im 8192]
constexpr size_t OFF_Y    = OFF_V    + (size_t)B*2*N*n;          // 256 * [re|im] 8192 each
constexpr size_t OFF_BM   = OFF_Y    + (size_t)2*B*2*N*n;        // 256 * [re|im] 4096 each
constexpr size_t OFF_D    = OFF_BM   + (size_t)2*B*2*n*n;
constexpr size_t OFF_X    = OFF_D    + (size_t)2*B*2*n*n;
constexpr size_t OFF_S    = OFF_X    + (size_t)2*B*2*n*n;
constexpr size_t OFF_PFS  = OFF_S    + (size_t)2*B*2*n*n;        // 256 * 4
}

// ---------------------------------------------------------------------------
// Tensor Data Mover: 2D tile (rows x cols, f32, row-major) global -> LDS.
// D# bit layout per CDNA5 ISA ch.8: group0 = {count, lds_addr, global_addr,
// type=2}; group1 = {data_size, tensor_dim0/1, tile_dim0/1, dim0_stride}.
// ---------------------------------------------------------------------------
#ifdef HAVE_TDM
typedef __attribute__((ext_vector_type(4))) unsigned int u32x4;
typedef __attribute__((ext_vector_type(4))) int i32x4;
typedef __attribute__((ext_vector_type(8))) int i32x8;

__device__ inline void tdm_load_2d(void* ldsDst, const float* gsrc,
                                   unsigned rows, unsigned cols,
                                   unsigned rowStride) {
  unsigned long long ga = (unsigned long long)(uintptr_t)gsrc;
  unsigned lds = (unsigned)(uintptr_t)ldsDst;  // flat LDS addr low 32 = offset
  u32x4 g0;
  g0.x = 1u;                                            // count=1 (valid D#)
  g0.y = lds;                                           // lds_addr [63:32]
  g0.z = (unsigned)(ga & 0xffffffffu);                  // global_addr[31:0]
  g0.w = (unsigned)((ga >> 32) & 0x01ffffffu) | 0x80000000u;  // [56:32]|type=2
  i32x8 g1;
  g1[0] = (int)(2u << 16);                              // data_size=4B, mask=0
  g1[1] = (int)((cols & 0xffffu) << 16);                // tensor_dim0 [63:48]
  g1[2] = (int)(((cols >> 16) & 0xffffu) |              // tensor_dim0 [79:64]
                ((rows & 0xffffu) << 16));              // tensor_dim1 [95:80]
  g1[3] = (int)(((rows >> 16) & 0xffffu) |              // tensor_dim1 [111:96]
                ((cols & 0xffffu) << 16));              // tile_dim0  [127:112]
  g1[4] = (int)(rows & 0xffffu);                        // tile_dim1; tile_dim2=0
  g1[5] = (int)rowStride;                               // dim0_stride low32
  g1[6] = 0;                                            // stride hi / dim1_stride
  g1[7] = 0;
  i32x4 z4 = {0, 0, 0, 0};
#if __clang_major__ >= 23
  i32x8 z8 = {0, 0, 0, 0, 0, 0, 0, 0};
  __builtin_amdgcn_tensor_load_to_lds(g0, g1, z4, z4, z8, 0);
#else
  __builtin_amdgcn_tensor_load_to_lds(g0, g1, z4, z4, 0);
#endif
}
#endif

// ---------------------------------------------------------------------------
// WMMA f32 16x16x4 tile machinery (wave32: 1 wave computes one 16x16 f32 tile)
// A-frag: lane L holds A[m][k0],A[m][k0+1], m=L%16, k0=kk+2*(L/16)
// B-frag: lane L holds B[k0][n],B[k0+1][n], n=L%16
// C/D:    lane L, vgpr r -> C[r + 8*(L/16)][L%16]
// ---------------------------------------------------------------------------
__device__ inline v2f ldA(const float* a, int aRS, int aCS, int m0, int k0) {
  int lane = threadIdx.x & 31;
  int m = m0 + (lane & 15);
  int k = k0 + ((lane >> 4) << 1);
  v2f r;
  r.x = a[(size_t)m * aRS + (size_t)k * aCS];
  r.y = a[(size_t)m * aRS + (size_t)(k + 1) * aCS];
  return r;
}
__device__ inline v2f ldB(const float* b, int bRS, int bCS, int k0, int n0) {
  int lane = threadIdx.x & 31;
  int nn = n0 + (lane & 15);
  int k = k0 + ((lane >> 4) << 1);
  v2f r;
  r.x = b[(size_t)k * bRS + (size_t)nn * bCS];
  r.y = b[(size_t)(k + 1) * bRS + (size_t)nn * bCS];
  return r;
}
__device__ inline v8f wmma4(v2f a, v2f b, v8f c) {
  // f32 WMMA has no A/B neg modifiers (NEG = {CNeg,0,0}); all mods zero.
  return __builtin_amdgcn_wmma_f32_16x16x4_f32(false, a, false, b, (short)0, c,
                                               false, false);
}
__device__ inline void stC(float* c, int cRS, int m0, int n0, v8f acc) {
  int lane = threadIdx.x & 31;
  int nn = n0 + (lane & 15);
  int mb = m0 + ((lane >> 4) << 3);
#pragma unroll
  for (int r2 = 0; r2 < 8; ++r2) c[(size_t)(mb + r2) * cRS + nn] = acc[r2];
}
__device__ inline v8f ldC(const float* c, int cRS, int m0, int n0) {
  v8f acc;
  int lane = threadIdx.x & 31;
  int nn = n0 + (lane & 15);
  int mb = m0 + ((lane >> 4) << 3);
#pragma unroll
  for (int r2 = 0; r2 < 8; ++r2) acc[r2] = c[(size_t)(mb + r2) * cRS + nn];
  return acc;
}

__device__ void mm_real(const float* A, int aRS, int aCS, const float* Bm,
                        int bRS, int bCS, float* C, int cRS, int K, int tm,
                        int tn) {
  v8f acc = {};
  for (int k = 0; k < K; k += 4) {
    __builtin_prefetch(&A[(size_t)tm * aRS + (size_t)(k + 16) * aCS], 0, 1);
    v2f a = ldA(A, aRS, aCS, tm, k);
    v2f b = ldB(Bm, bRS, bCS, k, tn);
    acc = wmma4(a, b, acc);
  }
  stC(C, cRS, tm, tn, acc);
}

// LDS-sourced variant (no prefetch; fragments come from ds_load)
__device__ void mm_real_lds(const float* A, int aRS, const float* Bm, int bRS,
                            float* C, int cRS, int K, int tm, int tn) {
  v8f acc = {};
  for (int k = 0; k < K; k += 4) {
    v2f a = ldA(A, aRS, 1, tm, k);
    v2f b = ldB(Bm, bRS, 1, k, tn);
    acc = wmma4(a, b, acc);
  }
  stC(C, cRS, tm, tn, acc);
}

// complex C = op(A)*op(B) (+ Dinit).  -aim*bim via VALU-negated A-imag frag.
__device__ void mm_cplx(const float* are, const float* aim, int aRS, int aCS,
                        const float* bre, const float* bim, int bRS, int bCS,
                        const float* dre, const float* dim_, float* cre,
                        float* cim, int cRS, int K, int tm, int tn) {
  v8f accr = {}, acci = {};
  if (dre) {
    accr = ldC(dre, cRS, tm, tn);
    acci = ldC(dim_, cRS, tm, tn);
  }
  for (int k = 0; k < K; k += 4) {
    v2f ar = ldA(are, aRS, aCS, tm, k);
    v2f ai = ldA(aim, aRS, aCS, tm, k);
    v2f br = ldB(bre, bRS, bCS, k, tn);
    v2f bi = ldB(bim, bRS, bCS, k, tn);
    v2f nai = -ai;                      // f32 WMMA lacks neg_a; negate in VALU
    accr = wmma4(ar, br, accr);
    accr = wmma4(nai, bi, accr);        // re -= aim*bim
    acci = wmma4(ar, bi, acci);
    acci = wmma4(ai, br, acci);
  }
  stC(cre, cRS, tm, tn, accr);
  stC(cim, cRS, tm, tn, acci);
}

// ---------------------------------------------------------------------------
// Stage kernels
// ---------------------------------------------------------------------------
__global__ void k_init(float* ws) {
  int t = threadIdx.x;
  if (t < 10) ws[gs::OFF_TR + t] = 0.f;
}

// H_g = skew_from_vec(raw_g), f64 -> f32
__global__ void k_expand(float* ws, const double* h1, const double* h2) {
  int idx = blockIdx.x * 256 + threadIdx.x;
  if (idx >= 2 * 16384) return;
  int g = idx >> 14, r = idx & 16383, i = r >> 7, j = r & 127;
  const double* v = g ? h2 : h1;
  float val = 0.f;
  if (i < j)
    val = (float)v[(size_t)i * (255 - i) / 2 + (j - i - 1)];
  else if (i > j)
    val = -(float)v[(size_t)j * (255 - j) / 2 + (i - j - 1)];
  ws[gs::OFF_H + idx] = val;
}

// generic 128x128x128 real matmul, batched over g
__global__ void k_mm128(float* ws, size_t aOff, size_t bOff, size_t cOff) {
  int g = blockIdx.y;
  int t = blockIdx.x;
  int tm = (t >> 3) << 4, tn = (t & 7) << 4;
  mm_real(ws + aOff + (size_t)g * 16384, 128, 1,
          ws + bOff + (size_t)g * 16384, 128, 1,
          ws + cOff + (size_t)g * 16384, 128, 128, tm, tn);
}

// G = -tanh(H/2) = -H/2 + H^3/24 - H^5/240 ; trace accumulation for log-trace
__global__ void k_green(float* ws) {
  int idx = blockIdx.x * 256 + threadIdx.x;
  if (idx >= 2 * 16384) return;
  int g = idx >> 14, r = idx & 16383, i = r >> 7, j = r & 127;
  size_t o = (size_t)g * 16384 + r;
  float h = ws[gs::OFF_H + o], p3 = ws[gs::OFF_P3 + o], p5 = ws[gs::OFF_P5 + o];
  ws[gs::OFF_G + o] = -0.5f * h + p3 * (1.f / 24.f) - p5 * (1.f / 240.f);
  float p2 = ws[gs::OFF_P2 + o];
  size_t ot = (size_t)g * 16384 + (size_t)j * 128 + i;
  if (i == j) atomicAdd(&ws[gs::OFF_TR + g * 3 + 0], p2);
  atomicAdd(&ws[gs::OFF_TR + g * 3 + 1], p2 * ws[gs::OFF_P2 + ot]);
  atomicAdd(&ws[gs::OFF_TR + g * 3 + 2], p3 * ws[gs::OFF_P3 + ot]);
}

// lt_g = 0.5*tr(log cosh(H/2)) = 0.5*(tr2/8 - tr4/192 + tr6/2880)
__global__ void k_lt(float* ws) {
  int t = threadIdx.x;
  if (t < 2) {
    float t2 = ws[gs::OFF_TR + t * 3 + 0];
    float t4 = ws[gs::OFF_TR + t * 3 + 1];
    float t6 = ws[gs::OFF_TR + t * 3 + 2];
    ws[gs::OFF_LT + t] =
        0.5f * (t2 * (1.f / 8.f) - t4 * (1.f / 192.f) + t6 * (1.f / 2880.f));
  }
}

// A_g = plus^T G plus (sparse gather; sample independent)
__global__ void k_buildA(float* ws) {
  int idx = blockIdx.x * 256 + threadIdx.x;
  if (idx >= 8192) return;
  int g = idx >> 12, r = idx & 4095, j = r >> 6, k = r & 63;
  const float* G = ws + gs::OFF_G + (size_t)g * 16384;
  float g00 = G[(2 * j) * 128 + 2 * k];
  float g01 = G[(2 * j) * 128 + 2 * k + 1];
  float g10 = G[(2 * j + 1) * 128 + 2 * k];
  float g11 = G[(2 * j + 1) * 128 + 2 * k + 1];
  float* A = ws + gs::OFF_A + (size_t)g * 8192;
  A[r] = 0.5f * (-g00 + g11);
  A[4096 + r] = -0.5f * (g01 + g10);
}

// V = vT^T: analytic solve of cyclic-bidiagonal M x = sp^H (per sample/col)
__global__ void k_buildV(float* ws, const double* x) {
  __shared__ float sg[64];
  int s = blockIdx.x, i = threadIdx.x;
  if (i < 64) {
    double a = x[(size_t)s * 64 + i], b = x[(size_t)s * 64 + ((i + 1) & 63)];
    float v = (a * b >= 0.0) ? 1.f : -1.f;
    if (i == 63) v = -v;  // PX = 1 boundary flip
    sg[i] = v;
  }
  __syncthreads();
  const float is2 = 0.70710678118f;
  float* vr = ws + gs::OFF_V + (size_t)s * 16384;
  float* vi = vr + 8192;
  int jstar;
  float cr, ci;
  if (i & 1) {  // odd memory row -> x_{j*} = 1/sqrt2
    jstar = (i - 1) >> 1;
    cr = is2; ci = 0.f;
  } else {      // even row -> x_{j*} = -i*sg/sqrt2
    jstar = ((i >> 1) + 63) & 63;
    cr = 0.f; ci = -sg[jstar] * is2;
  }
  for (int tt = 0; tt < 64; ++tt) {
    int j = (jstar - tt + 64) & 63;
    if (tt > 0) { cr *= sg[j]; ci *= sg[j]; }
    vr[(size_t)i * 64 + j] = cr;
    vi[(size_t)i * 64 + j] = ci;
  }
}

// Y_{g,s,p} = G_g @ V_{s,p}. One workgroup per (g,s,plane): TDM-stage G (64KB)
// and V plane (32KB) into LDS, then 8 waves x 4 WMMA tiles consume via ds_load.
__global__ void k_mmY(float* ws) {
  __shared__ float Gs[128 * 128];   // 64 KB
  __shared__ float Vs[128 * 64];    // 32 KB
  int z = blockIdx.x;               // ((g*128+s)*2 + plane)
  int p = z & 1, gz = z >> 1, g = gz >> 7, s = gz & 127;
  const float* Gp = ws + gs::OFF_G + (size_t)g * 16384;
  const float* Vp = ws + gs::OFF_V + (size_t)s * 16384 + (size_t)p * 8192;
  float* Yp = ws + gs::OFF_Y + (size_t)gz * 16384 + (size_t)p * 8192;
#ifdef HAVE_TDM
  if (threadIdx.x < 32) {  // wave 0 issues both tensor DMAs (EXEC-independent)
    tdm_load_2d(Gs, Gp, 128, 128, 128);
    tdm_load_2d(Vs, Vp, 128, 64, 64);
    __builtin_amdgcn_s_wait_tensorcnt(0);
  }
  __syncthreads();
#else
  for (int idx = threadIdx.x; idx < 128 * 128; idx += 256) Gs[idx] = Gp[idx];
  for (int idx = threadIdx.x; idx < 128 * 64; idx += 256) Vs[idx] = Vp[idx];
  __syncthreads();
#endif
  int wave = threadIdx.x >> 5;
#pragma unroll
  for (int tt = 0; tt < 4; ++tt) {
    int tile = wave * 4 + tt;        // 32 tiles over 128x64 output
    int tm = (tile >> 2) << 4, tn = (tile & 3) << 4;
    mm_real_lds(Gs, 128, Vs, 64, Yp, 64, 128, tm, tn);
  }
}

// B = -I + u^T Y, with u = -plus (sparse gather)
__global__ void k_buildB(float* ws) {
  int idx = blockIdx.x * 256 + threadIdx.x;
  int z = idx >> 12, r = idx & 4095, j = r >> 6, k = r & 63;
  const float* yre = ws + gs::OFF_Y + (size_t)z * 16384;
  const float* yim = yre + 8192;
  const float is2 = 0.70710678118f;
  float br = (-yim[(2 * j) * 64 + k] - yre[(2 * j + 1) * 64 + k]) * is2 -
             ((j == k) ? 1.f : 0.f);
  float bi = (yre[(2 * j) * 64 + k] - yim[(2 * j + 1) * 64 + k]) * is2;
  float* b = ws + gs::OFF_BM + (size_t)z * 8192;
  b[r] = br;
  b[4096 + r] = bi;
}

// D = V^T Y (complex 64x128x64)
__global__ void k_mmD(float* ws) {
  int z = blockIdx.y, s = z & 127;
  int t = blockIdx.x;
  int tm = (t >> 2) << 4, tn = (t & 3) << 4;
  const float* vr = ws + gs::OFF_V + (size_t)s * 16384;
  const float* yr = ws + gs::OFF_Y + (size_t)z * 16384;
  float* dr = ws + gs::OFF_D + (size_t)z * 8192;
  mm_cplx(vr, vr + 8192, 1, 64,          // V^T via strides
          yr, yr + 8192, 64, 1, nullptr, nullptr, dr, dr + 4096, 64, 128, tm,
          tn);
}

// X = Ainv_g @ B_{g,s}
__global__ void k_mmX(float* ws) {
  int z = blockIdx.y, g = z >> 7;
  int t = blockIdx.x;
  int tm = (t >> 2) << 4, tn = (t & 3) << 4;
  const float* ar = ws + gs::OFF_AINV + (size_t)g * 8192;
  const float* br = ws + gs::OFF_BM + (size_t)z * 8192;
  float* xr = ws + gs::OFF_X + (size_t)z * 8192;
  mm_cplx(ar, ar + 4096, 64, 1, br, br + 4096, 64, 1, nullptr, nullptr, xr,
          xr + 4096, 64, 64, tm, tn);
}

// S = D + B^T X
__global__ void k_mmS(float* ws) {
  int z = blockIdx.y;
  int t = blockIdx.x;
  int tm = (t >> 2) << 4, tn = (t & 3) << 4;
  const float* br = ws + gs::OFF_BM + (size_t)z * 8192;
  const float* xr = ws + gs::OFF_X + (size_t)z * 8192;
  const float* dr = ws + gs::OFF_D + (size_t)z * 8192;
  float* sr = ws + gs::OFF_S + (size_t)z * 8192;
  mm_cplx(br, br + 4096, 1, 64,          // B^T via strides
          xr, xr + 4096, 64, 1, dr, dr + 4096, sr, sr + 4096, 64, 64, tm, tn);
}

// ---------------------------------------------------------------------------
// Cooperative 64x64 complex Pfaffian (Householder), matrix resident in LDS
// ---------------------------------------------------------------------------
__global__ void k_pfaffian64(float* ws, size_t matOff, size_t matStride,
                             size_t outOff) {
  __shared__ float mre[64 * 64], mim[64 * 64];
  __shared__ float vre[64], vim[64], wre_s[64], wim_s[64], red[64];
  __shared__ float sc[8];   // vn.re/|v|, vn.im/|v|, 1/|v|, tau, al.re, al.im, sigma
  __shared__ float acc_s[4];  // sgn.re, sgn.im, logpf
  int t = threadIdx.x;
  const float* src = ws + matOff + (size_t)blockIdx.x * matStride;
  for (int c = 0; c < 64; ++c) {
    mre[t * 64 + c] = src[t * 64 + c];
    mim[t * 64 + c] = src[4096 + t * 64 + c];
  }
  if (t == 0) { acc_s[0] = 1.f; acc_s[1] = 0.f; acc_s[2] = 0.f; }
  __syncthreads();
  for (int i = 0; i < 62; ++i) {
    int nn = i + 1;
    float xr = mre[t * 64 + i], xi = mim[t * 64 + i];
    red[t] = (t > nn) ? (xr * xr + xi * xi) : 0.f;
    __syncthreads();
    if (t == 0) {
      float sigma = 0.f;
      for (int r = 0; r < 64; ++r) sigma += red[r];
      float xnr = mre[nn * 64 + i], xni = mim[nn * 64 + i];
      float ax2 = xnr * xnr + xni * xni;
      float nx = sqrtf(ax2 + sigma), ax = sqrtf(ax2);
      float phr = 1.f, phi = 0.f;
      if (ax != 0.f) { phr = xnr / ax; phi = xni / ax; }
      float vnr = xnr + phr * nx, vni = xni + phi * nx;
      float alr = -phr * nx, ali = -phi * nx, tau = 2.f;
      float vnrm = sqrtf(vnr * vnr + vni * vni + sigma);
      if (sigma == 0.f) { tau = 0.f; alr = xnr; ali = xni; vnrm = 1.f; vnr = 0.f; vni = 0.f; }
      sc[0] = vnr / vnrm; sc[1] = vni / vnrm; sc[2] = 1.f / vnrm;
      sc[3] = tau; sc[4] = alr; sc[5] = ali; sc[6] = sigma;
    }
    __syncthreads();
    float tau = sc[3], vr, vi;
    if (sc[6] == 0.f) { vr = 0.f; vi = 0.f; }
    else if (t == nn) { vr = sc[0]; vi = sc[1]; }
    else if (t > nn) { vr = xr * sc[2]; vi = xi * sc[2]; }
    else { vr = 0.f; vi = 0.f; }
    vre[t] = vr; vim[t] = vi;
    __syncthreads();
    float wr = 0.f, wi = 0.f;          // w = tau * A @ conj(v)
    for (int c = nn; c < 64; ++c) {
      float ar = mre[t * 64 + c], ai = mim[t * 64 + c];
      float cr = vre[c], ci = -vim[c];
      wr += ar * cr - ai * ci;
      wi += ar * ci + ai * cr;
    }
    wre_s[t] = tau * wr; wim_s[t] = tau * wi;
    __syncthreads();
    float vtr = vre[t], vti = vim[t], wtr = wre_s[t], wti = wim_s[t];
    for (int c = 0; c < 64; ++c) {     // A += outer(v,w) - outer(w,v)
      float addr = vtr * wre_s[c] - vti * wim_s[c] - (wtr * vre[c] - wti * vim[c]);
      float addi = vtr * wim_s[c] + vti * wre_s[c] - (wtr * vim[c] + wti * vre[c]);
      mre[t * 64 + c] += addr;
      mim[t * 64 + c] += addi;
    }
    if (t == 0) {
      float omt = 1.f - tau;           // real
      float l = acc_s[2] + logf(fmaxf(fabsf(omt), 1e-30f));
      float sr = acc_s[0], si = acc_s[1];
      float f = (omt >= 0.f) ? 1.f : -1.f;
      sr *= f; si *= f;
      if ((i & 1) == 0) {
        float ar = -sc[4], ai = -sc[5];
        float aa = sqrtf(ar * ar + ai * ai);
        l += logf(fmaxf(aa, 1e-30f));
        float pr = ar / aa, pi = ai / aa;
        float nsr = sr * pr - si * pi, nsi = sr * pi + si * pr;
        sr = nsr; si = nsi;
      }
      acc_s[0] = sr; acc_s[1] = si; acc_s[2] = l;
    }
    __syncthreads();
  }
  if (t == 0) {
    float lr = mre[62 * 64 + 63], li = mim[62 * 64 + 63];
    float aa = sqrtf(lr * lr + li * li);
    float l = acc_s[2] + logf(fmaxf(aa, 1e-30f));
    float sr = acc_s[0] * (lr / aa) - acc_s[1] * (li / aa);
    float si = acc_s[0] * (li / aa) + acc_s[1] * (lr / aa);
    float* o = ws + outOff + (size_t)blockIdx.x * 4;
    o[0] = sr; o[1] = si; o[2] = l;
  }
}

// Cooperative Gauss-Jordan inversion of 64x64 complex A (pivoted), A in LDS
__global__ void k_invert64(float* ws, size_t matOff, size_t matStride,
                           size_t invOff, size_t invStride) {
  __shared__ float mre[64 * 64], mim[64 * 64];
  __shared__ float sc[2];
  __shared__ int prow_s;
  int t = threadIdx.x;
  const float* src = ws + matOff + (size_t)blockIdx.x * matStride;
  float* inv = ws + invOff + (size_t)blockIdx.x * invStride;
  for (int c = 0; c < 64; ++c) {
    mre[t * 64 + c] = src[t * 64 + c];
    mim[t * 64 + c] = src[4096 + t * 64 + c];
    inv[t * 64 + c] = (t == c) ? 1.f : 0.f;
    inv[4096 + t * 64 + c] = 0.f;
  }
  __syncthreads();
  __threadfence_block();
  for (int p = 0; p < 64; ++p) {
    if (t == 0) {
      int pr = p;
      float best = mre[p * 64 + p] * mre[p * 64 + p] + mim[p * 64 + p] * mim[p * 64 + p];
      for (int r = p + 1; r < 64; ++r) {
        float v = mre[r * 64 + p] * mre[r * 64 + p] + mim[r * 64 + p] * mim[r * 64 + p];
        if (v > best) { best = v; pr = r; }
      }
      prow_s = pr;
    }
    __syncthreads();
    int pr = prow_s;
    if (pr != p) {  // swap rows p<->pr; thread t handles column t
      float a;
      a = mre[p * 64 + t]; mre[p * 64 + t] = mre[pr * 64 + t]; mre[pr * 64 + t] = a;
      a = mim[p * 64 + t]; mim[p * 64 + t] = mim[pr * 64 + t]; mim[pr * 64 + t] = a;
      a = inv[p * 64 + t]; inv[p * 64 + t] = inv[pr * 64 + t]; inv[pr * 64 + t] = a;
      a = inv[4096 + p * 64 + t]; inv[4096 + p * 64 + t] = inv[4096 + pr * 64 + t]; inv[4096 + pr * 64 + t] = a;
    }
    __syncthreads();
    __threadfence_block();
    if (t == 0) {
      float dr = mre[p * 64 + p], di = mim[p * 64 + p];
      float d2 = dr * dr + di * di;
      if (d2 == 0.f) d2 = 1e-30f;
      sc[0] = dr / d2; sc[1] = -di / d2;
    }
    __syncthreads();
    {  // scale pivot row; thread t handles column t
      float ir = sc[0], ii = sc[1];
      float ar = mre[p * 64 + t], ai = mim[p * 64 + t];
      mre[p * 64 + t] = ar * ir - ai * ii;
      mim[p * 64 + t] = ar * ii + ai * ir;
      ar = inv[p * 64 + t]; ai = inv[4096 + p * 64 + t];
      inv[p * 64 + t] = ar * ir - ai * ii;
      inv[4096 + p * 64 + t] = ar * ii + ai * ir;
    }
    __syncthreads();
    __threadfence_block();
    if (t != p) {  // eliminate; thread t = row
      float fr = mre[t * 64 + p], fi = mim[t * 64 + p];
      for (int c = 0; c < 64; ++c) {
        float pvr = mre[p * 64 + c], pvi = mim[p * 64 + c];
        mre[t * 64 + c] -= fr * pvr - fi * pvi;
        mim[t * 64 + c] -= fr * pvi + fi * pvr;
        pvr = inv[p * 64 + c]; pvi = inv[4096 + p * 64 + c];
        inv[t * 64 + c] -= fr * pvr - fi * pvi;
        inv[4096 + t * 64 + c] -= fr * pvi + fi * pvr;
      }
    }
    __syncthreads();
    __threadfence_block();
  }
}

// lo_g = lA + lS + lt + i*arg(sA*sS) ; out = complex logsumexp(lo1, log(sprod)+lo2)
__global__ void k_combine(float* ws, const double* x, float* out, int out_size) {
  int s = threadIdx.x;
  if (s >= 128) return;
  float sp = 1.f;
  for (int j = 0; j < 64; ++j)
    sp *= (x[(size_t)s * 64 + j] >= 0.0) ? 1.f : -1.f;
  float re[2], ph[2];
  for (int g = 0; g < 2; ++g) {
    float sAr = ws[gs::OFF_PFA + g * 4], sAi = ws[gs::OFF_PFA + g * 4 + 1];
    float lA = ws[gs::OFF_PFA + g * 4 + 2];
    int z = g * 128 + s;
    float sSr = ws[gs::OFF_PFS + z * 4], sSi = ws[gs::OFF_PFS + z * 4 + 1];
    float lS = ws[gs::OFF_PFS + z * 4 + 2];
    float pr = sAr * sSr - sAi * sSi, pi = sAr * sSi + sAi * sSr;
    re[g] = lA + lS + ws[gs::OFF_LT + g];
    ph[g] = atan2f(pi, pr);
  }
  if (sp < 0.f) ph[1] += 3.14159265358979f;
  float m = fmaxf(re[0], re[1]);
  float e0 = expf(re[0] - m), e1 = expf(re[1] - m);
  float ar = e0 * cosf(ph[0]) + e1 * cosf(ph[1]);
  float ai = e0 * sinf(ph[0]) + e1 * sinf(ph[1]);
  float outre = m + 0.5f * logf(ar * ar + ai * ai);
  float outim = atan2f(ai, ar);
  if (out_size >= 256) {
    out[2 * s] = outre;
    out[2 * s + 1] = outim;
  } else if (s < out_size) {
    out[s] = outre;
  }
}

// ---------------------------------------------------------------------------
extern "C" void kernel_launch(void* const* d_in, const int* in_sizes, int n_in,
                              void* d_out, int out_size, void* d_ws,
                              size_t ws_size, hipStream_t stream) {
  (void)in_sizes; (void)n_in; (void)ws_size;
  const double* x = (const double*)d_in[0];
  const double* h1 = (const double*)d_in[1];
  const double* h2 = (const double*)d_in[2];
  float* ws = (float*)d_ws;
  float* out = (float*)d_out;

  k_init<<<1, 32, 0, stream>>>(ws);
  k_expand<<<128, 256, 0, stream>>>(ws, h1, h2);
  // matrix polynomial for G and traces: P2=H*H, P3=P2*H, P5=P3*P2
  k_mm128<<<dim3(64, 2), 32, 0, stream>>>(ws, gs::OFF_H, gs::OFF_H, gs::OFF_P2);
  k_mm128<<<dim3(64, 2), 32, 0, stream>>>(ws, gs::OFF_P2, gs::OFF_H, gs::OFF_P3);
  k_mm128<<<dim3(64, 2), 32, 0, stream>>>(ws, gs::OFF_P3, gs::OFF_P2, gs::OFF_P5);
  k_green<<<128, 256, 0, stream>>>(ws);
  k_lt<<<1, 32, 0, stream>>>(ws);
  k_buildA<<<32, 256, 0, stream>>>(ws);
  k_pfaffian64<<<2, 64, 0, stream>>>(ws, gs::OFF_A, 8192, gs::OFF_PFA);
  k_invert64<<<2, 64, 0, stream>>>(ws, gs::OFF_A, 8192, gs::OFF_AINV, 8192);
  k_buildV<<<128, 128, 0, stream>>>(ws, x);
  k_mmY<<<512, 256, 0, stream>>>(ws);
  k_buildB<<<4096, 256, 0, stream>>>(ws);
  k_mmD<<<dim3(16, 256), 32, 0, stream>>>(ws);
  k_mmX<<<dim3(16, 256), 32, 0, stream>>>(ws);
  k_mmS<<<dim3(16, 256), 32, 0, stream>>>(ws);
  k_pfaffian64<<<256, 64, 0, stream>>>(ws, gs::OFF_S, 8192, gs::OFF_PFS);
  k_combine<<<1, 128, 0, stream>>>(ws, x, out, out_size);
}